// ContrastiveCenterLoss_11450382812046
// MI455X (gfx1250) — compile-verified
//
#include <hip/hip_runtime.h>

typedef __attribute__((ext_vector_type(2))) float v2f;
typedef __attribute__((ext_vector_type(8))) float v8f;

#define BATCH   2048
#define NCLS    100
#define FDIM    512
#define NTILE_N 7                     // ceil(100/16)
#define NTILE_M (BATCH / 16)          // 128
#define NTILES  (NTILE_M * NTILE_N)   // 896

#define LAMBDA_C 1.0f
#define EPSILON  1e-6f

// ---------------------------------------------------------------------------
// Kernel 1: per-row squared norms for feat (2048 rows) and weight (100 rows).
// One wave32 per row; float4 coalesced loads; shuffle reduction.
// ---------------------------------------------------------------------------
__global__ __launch_bounds__(256)
void ccl_rownorm_kernel(const float* __restrict__ feat,
                        const float* __restrict__ weight,
                        float* __restrict__ f2, float* __restrict__ c2) {
  int wave = (int)(blockIdx.x * (blockDim.x / 32u) + (threadIdx.x >> 5));
  int lane = (int)(threadIdx.x & 31u);
  if (wave >= BATCH + NCLS) return;

  const float* row = (wave < BATCH) ? (feat + (size_t)wave * FDIM)
                                    : (weight + (size_t)(wave - BATCH) * FDIM);
  const float4* r4 = (const float4*)row;    // 128 float4 per row
  float s = 0.0f;
#pragma unroll
  for (int j = 0; j < 4; ++j) {
    float4 v = r4[lane + j * 32];
    s += v.x * v.x + v.y * v.y + v.z * v.z + v.w * v.w;
  }
#pragma unroll
  for (int off = 16; off > 0; off >>= 1) s += __shfl_down(s, off, 32);
  if (lane == 0) {
    if (wave < BATCH) f2[wave] = s;
    else              c2[wave - BATCH] = s;
  }
}

// ---------------------------------------------------------------------------
// Kernel 2: one wave per 16x16 tile of cross = feat @ weight^T, computed with
// V_WMMA_F32_16X16X4_F32 over K=512 (128 WMMAs). Epilogue folds f2/c2 in and
// produces per-tile (total_sum, intra_sum) partials.
//
// A 16x4 f32 layout (ISA 7.12.2): lane L holds row M=L&15; VGPR0/1 = K 0/1
// for L<16, K 2/3 for L>=16.  B 4x16 mirrors with N=L&15.  C/D 16x16 f32:
// element (m,n) at vgpr v, lane L with m = v + 8*(L>>4), n = L&15.
// ---------------------------------------------------------------------------
__global__ __launch_bounds__(32)
void ccl_tile_wmma_kernel(const float* __restrict__ feat,
                          const float* __restrict__ weight,
                          const int*   __restrict__ label,
                          const float* __restrict__ f2,
                          const float* __restrict__ c2,
                          float* __restrict__ partials) {
  const int tile = (int)blockIdx.x;        // 0..895
  const int tm   = tile / NTILE_N;         // M tile  0..127
  const int tn   = tile % NTILE_N;         // N tile  0..6
  const int lane = (int)(threadIdx.x & 31u);
  const int hb   = lane >> 4;              // half-wave: 0 or 1
  const int l15  = lane & 15;

  const int mrow = tm * 16 + l15;                      // feat row (always valid)
  const int ncol = tn * 16 + l15;                      // logical class column
  const int nrow = (ncol < NCLS) ? ncol : (NCLS - 1);  // clamped for safe loads

  const float* arow = feat   + (size_t)mrow * FDIM + hb * 2;
  const float* brow = weight + (size_t)nrow * FDIM + hb * 2;
  __builtin_prefetch(brow, 0, 3);          // global_prefetch_b8

  v8f acc = {};
#pragma unroll 8
  for (int k = 0; k < FDIM; k += 4) {
    v2f a = *(const v2f*)(arow + k);
    v2f b = *(const v2f*)(brow + k);
    // D = A(16x4) * B(4x16) + C ; 8 args: neg_a, A, neg_b, B, c_mod, C, reuse_a, reuse_b
    acc = __builtin_amdgcn_wmma_f32_16x16x4_f32(false, a, false, b,
                                                (short)0, acc, false, false);
  }

  // Epilogue: d[m][n] = f2[m] + c2[n] - 2*cross[m][n], masked for n >= 100.
  float tot = 0.0f, intra = 0.0f;
  if (ncol < NCLS) {
    const float c2n = c2[ncol];
#pragma unroll
    for (int v = 0; v < 8; ++v) {
      int m = tm * 16 + v + hb * 8;
      float d = f2[m] + c2n - 2.0f * acc[v];
      tot += d;
      if (label[m] == ncol) intra += d;
    }
  }
#pragma unroll
  for (int off = 16; off > 0; off >>= 1) {
    tot   += __shfl_down(tot,   off, 32);
    intra += __shfl_down(intra, off, 32);
  }
  if (lane == 0) {
    partials[2 * tile + 0] = tot;
    partials[2 * tile + 1] = intra;
  }
}

// ---------------------------------------------------------------------------
// Kernel 3: deterministic tree reduction of the 896 tile partials + scalar
// loss epilogue in the reference's exact operation order.
// ---------------------------------------------------------------------------
__global__ __launch_bounds__(256)
void ccl_finalize_kernel(const float* __restrict__ partials,
                         float* __restrict__ out) {
  __shared__ float st[256];
  __shared__ float si[256];
  int t = (int)threadIdx.x;
  float tot = 0.0f, intra = 0.0f;
  for (int i = t; i < NTILES; i += 256) {
    tot   += partials[2 * i + 0];
    intra += partials[2 * i + 1];
  }
  st[t] = tot; si[t] = intra;
  __syncthreads();
  for (int s = 128; s > 0; s >>= 1) {
    if (t < s) { st[t] += st[t + s]; si[t] += si[t + s]; }
    __syncthreads();
  }
  if (t == 0) {
    float intra_d = si[0];
    float inter_d = st[0] - intra_d;
    float loss = LAMBDA_C / 2.0f / (float)BATCH * intra_d / (inter_d + EPSILON) / 0.1f;
    out[0] = loss;
  }
}

// ---------------------------------------------------------------------------
extern "C" void kernel_launch(void* const* d_in, const int* in_sizes, int n_in,
                              void* d_out, int out_size, void* d_ws, size_t ws_size,
                              hipStream_t stream) {
  const float* feat   = (const float*)d_in[0];   // 2048*512
  const float* weight = (const float*)d_in[1];   // 100*512
  const int*   label  = (const int*)d_in[2];     // 2048

  float* ws       = (float*)d_ws;
  float* partials = ws;                  // 2 * 896
  float* f2       = ws + 2 * NTILES;     // 2048
  float* c2       = f2 + BATCH;          // 100

  // Kernel 1: 2148 waves -> 8 waves per 256-thread block -> 269 blocks
  int nrows   = BATCH + NCLS;
  int nblocks = (nrows + 7) / 8;
  ccl_rownorm_kernel<<<nblocks, 256, 0, stream>>>(feat, weight, f2, c2);

  // Kernel 2: one wave32 per 16x16 tile
  ccl_tile_wmma_kernel<<<NTILES, 32, 0, stream>>>(feat, weight, label, f2, c2, partials);

  // Kernel 3: single-block deterministic reduction + loss
  ccl_finalize_kernel<<<1, 256, 0, stream>>>(partials, (float*)d_out);
}